// NequIPMessagePassingLayer_8847632630532
// MI455X (gfx1250) — compile-verified
//
#include <hip/hip_runtime.h>
#include <math.h>

// ---------------------------------------------------------------------------
// NequIP message-passing layer for MI455X (gfx1250, wave32, WMMA).
// f32 WMMA 16x16x4 for all GEMMs; L2-resident scratch; hardware f32 atomics
// for the edge->node segment sum. B-matrix fragments are stored pre-paired
// (float2 {B[k][n], B[k+1][n]}) so each WMMA B-fragment is a single
// ds_load_b64 (compiler fuses pairs into ds_load_2addr_stride64_b64).
// Epilogues use tile-level bound checks so full tiles store branchlessly
// with immediate-offset global stores.
// ---------------------------------------------------------------------------

typedef __attribute__((ext_vector_type(2))) float v2f;
typedef __attribute__((ext_vector_type(8))) float v8f;

__device__ __forceinline__ v8f wmma4(v2f a, v2f b, v8f c) {
  // V_WMMA_F32_16X16X4_F32 : D(16x16,f32) = A(16x4,f32) x B(4x16,f32) + C
  return __builtin_amdgcn_wmma_f32_16x16x4_f32(
      /*neg_a=*/false, a, /*neg_b=*/false, b,
      /*c_mod=*/(short)0, c, /*reuse_a=*/false, /*reuse_b=*/false);
}

__device__ __forceinline__ float silu_f(float x) {
  return x / (1.0f + __expf(-x));
}

__device__ __forceinline__ void atomicAddF(float* p, float v) {
  // relaxed agent-scope -> global_atomic_add_f32 (no return, no CAS loop)
  __hip_atomic_fetch_add(p, v, __ATOMIC_RELAXED, __HIP_MEMORY_SCOPE_AGENT);
}

// ---------------------------------------------------------------------------
// Kernel 0: build fused / pre-scaled weight matrices in paired fragment order.
//
//  Wcs (as v2f): 9 chunks (nc=0..2 x kc=0..2), each 64x64 v2f (32KB):
//     chunk[nc*3+kc][p*64+col] = { W[kc*128+2p][nc*64+col],
//                                  W[kc*128+2p+1][nc*64+col] }
//     where W rows 0..127 = post_w_s/sqrt(128), rows 128..383 = si_w_s/16
//  Wcv (as v2f): 3 chunks (kc=0..2), each 64x64 v2f, same row structure with
//     post_w_v and si_w_v.
//  We  (as v2f): 32x64 v2f pairs of post_w_e/8.
// ---------------------------------------------------------------------------
__global__ void k_build_w(const float* __restrict__ pws,
                          const float* __restrict__ pwv,
                          const float* __restrict__ pwe,
                          const float* __restrict__ sis,
                          const float* __restrict__ siv,
                          v2f* __restrict__ Wcs,
                          v2f* __restrict__ Wcv,
                          v2f* __restrict__ We) {
  const float scPost = 0.08838834764831845f;  // 1/sqrt(128)
  const float scSi   = 0.0625f;               // 1/sqrt(64*4)
  int i = blockIdx.x * blockDim.x + threadIdx.x;
  if (i < 36864) {  // Wcs: 9 * 4096 v2f
    int chunk = i >> 12, within = i & 4095;
    int p = within >> 6, col = within & 63;
    int nc = chunk / 3, kc = chunk % 3;
    int k0 = kc * 128 + 2 * p;
    int v = nc * 64 + col;
    v2f val;
    val.x = (k0 < 128) ? pws[k0 * 192 + v] * scPost
                       : sis[(k0 - 128) * 192 + v] * scSi;
    val.y = (k0 + 1 < 128) ? pws[(k0 + 1) * 192 + v] * scPost
                           : sis[(k0 + 1 - 128) * 192 + v] * scSi;
    Wcs[i] = val;
  } else if (i < 36864 + 12288) {  // Wcv: 3 * 4096 v2f
    int j = i - 36864;
    int kc = j >> 12, within = j & 4095;
    int p = within >> 6, col = within & 63;
    int k0 = kc * 128 + 2 * p;
    v2f val;
    val.x = (k0 < 128) ? pwv[k0 * 64 + col] * scPost
                       : siv[(k0 - 128) * 64 + col] * scSi;
    val.y = (k0 + 1 < 128) ? pwv[(k0 + 1) * 64 + col] * scPost
                           : siv[(k0 + 1 - 128) * 64 + col] * scSi;
    Wcv[j] = val;
  } else if (i < 36864 + 12288 + 2048) {  // We: 2048 v2f
    int j = i - 36864 - 12288;
    int p = j >> 6, col = j & 63;
    v2f val;
    val.x = pwe[(2 * p) * 64 + col] * 0.125f;
    val.y = pwe[(2 * p + 1) * 64 + col] * 0.125f;
    We[j] = val;
  }
}

// ---------------------------------------------------------------------------
// Kernel 1: ps = s @ pw_s / 8 ; pv[:,:,c] = v[:,:,c] @ pw_v / 8
// One wave per 16-node tile. Paired-fragment weights staged in LDS.
// ---------------------------------------------------------------------------
__global__ __launch_bounds__(128) void k_node_proj(
    const float* __restrict__ emb, const float* __restrict__ pw_s,
    const float* __restrict__ pw_v, float* __restrict__ ps,
    float* __restrict__ pv, int N, int nTiles) {
  __shared__ v2f Bs2[32 * 64];
  __shared__ v2f Bv2[32 * 64];
  int t = threadIdx.x;
  for (int i = t; i < 32 * 64; i += 128) {
    int p = i >> 6, n = i & 63;
    v2f a, b;
    a.x = pw_s[(2 * p) * 64 + n]; a.y = pw_s[(2 * p + 1) * 64 + n];
    b.x = pw_v[(2 * p) * 64 + n]; b.y = pw_v[(2 * p + 1) * 64 + n];
    Bs2[i] = a; Bv2[i] = b;
  }
  __syncthreads();
  int wave = t >> 5, lane = t & 31;
  int tile = blockIdx.x * 4 + wave;
  if (tile >= nTiles) return;
  int m = lane & 15, kk = (lane >> 4) << 1, kh = lane >> 4;
  int node = tile * 16 + m; if (node > N - 1) node = N - 1;
  const float* srow = emb + (size_t)node * 256;       // s part
  const float* vrow = srow + 64;                      // v part, v[u][c]=vrow[u*3+c]

  int rbase = tile * 16 + (kh << 3);   // first of this lane-group's 8 rows
  bool full = (rbase + 8 <= N);

  for (int nt = 0; nt < 4; ++nt) {
    v8f acc = {};
    for (int kt = 0; kt < 16; ++kt) {
      int k = kt * 4 + kk;
      v2f a; a.x = srow[k]; a.y = srow[k + 1];
      v2f b = Bs2[(kt * 2 + kh) * 64 + nt * 16 + m];
      acc = wmma4(a, b, acc);
    }
    if (full) {
      float* dst = ps + (size_t)rbase * 64 + nt * 16 + (lane & 15);
#pragma unroll
      for (int rr = 0; rr < 8; ++rr) dst[(size_t)rr * 64] = acc[rr] * 0.125f;
    } else {
      for (int rr = 0; rr < 8; ++rr) {
        int nn = rbase + rr;
        if (nn < N) ps[(size_t)nn * 64 + nt * 16 + (lane & 15)] = acc[rr] * 0.125f;
      }
    }
  }
  for (int c = 0; c < 3; ++c) {
    for (int nt = 0; nt < 4; ++nt) {
      v8f acc = {};
      for (int kt = 0; kt < 16; ++kt) {
        int k = kt * 4 + kk;
        v2f a; a.x = vrow[k * 3 + c]; a.y = vrow[(k + 1) * 3 + c];
        v2f b = Bv2[(kt * 2 + kh) * 64 + nt * 16 + m];
        acc = wmma4(a, b, acc);
      }
      if (full) {
        float* dst = pv + (size_t)rbase * 192 + (nt * 16 + (lane & 15)) * 3 + c;
#pragma unroll
        for (int rr = 0; rr < 8; ++rr) dst[(size_t)rr * 192] = acc[rr] * 0.125f;
      } else {
        for (int rr = 0; rr < 8; ++rr) {
          int nn = rbase + rr;
          if (nn < N)
            pv[(size_t)nn * 192 + (nt * 16 + (lane & 15)) * 3 + c] = acc[rr] * 0.125f;
        }
      }
    }
  }
}

// ---------------------------------------------------------------------------
// Kernel 2: per-edge radial MLP + messages + atomic segment-sum.
// One edge per wave32; lane handles channels (lane, lane+32).
// ---------------------------------------------------------------------------
__global__ __launch_bounds__(256) void k_edges(
    const float* __restrict__ dist, const float* __restrict__ eemb,
    const int* __restrict__ cidx, const int* __restrict__ nidx,
    const float* __restrict__ bf,
    const float* __restrict__ W1, const float* __restrict__ b1,
    const float* __restrict__ W2, const float* __restrict__ b2,
    const float* __restrict__ W3, const float* __restrict__ b3,
    const float* __restrict__ ps, const float* __restrict__ pv,
    float* __restrict__ M0, float* __restrict__ M1o, float* __restrict__ M1e,
    int nEdges) {
  __shared__ float sW3[10 * 320];
  __shared__ float sB3[320];
  __shared__ float sW1[80];
  __shared__ float sW2[100];
  __shared__ float sB1[10], sB2[10], sBF[8];
  int t = threadIdx.x;
  for (int i = t; i < 3200; i += 256) sW3[i] = W3[i];
  for (int i = t; i < 320; i += 256) sB3[i] = b3[i];
  if (t < 80) sW1[t] = W1[t];
  if (t < 100) sW2[t] = W2[t];
  if (t < 10) { sB1[t] = b1[t]; sB2[t] = b2[t]; }
  if (t < 8) sBF[t] = bf[t];
  __syncthreads();

  int wave = t >> 5, lane = t & 31;
  int e = blockIdx.x * 8 + wave;
  if (e >= nEdges) return;

  float r = dist[e];
  int n = cidx[e], j = nidx[e];
  float e0  = eemb[e * 4 + 0];
  float e1x = eemb[e * 4 + 1], e1y = eemb[e * 4 + 2], e1z = eemb[e * 4 + 3];

  // polynomial envelope + Bessel radial basis
  float x = r * 0.2f;
  float x2 = x * x, x6 = x2 * x2 * x2, x7 = x6 * x, x8 = x7 * x;
  float env = (x < 1.0f) ? (1.0f - 28.0f * x6 + 48.0f * x7 - 21.0f * x8) : 0.0f;
  float pref = sqrtf(0.4f) * env / r;  // sqrt(2/CUTOFF)/r * env
  float rad[8];
#pragma unroll
  for (int i = 0; i < 8; ++i) rad[i] = pref * __sinf(sBF[i] * r);

  float h1[10];
#pragma unroll
  for (int o = 0; o < 10; ++o) {
    float a = sB1[o];
#pragma unroll
    for (int i = 0; i < 8; ++i) a += rad[i] * sW1[i * 10 + o];
    h1[o] = silu_f(a);
  }
  float h2[10];
#pragma unroll
  for (int o = 0; o < 10; ++o) {
    float a = sB2[o];
#pragma unroll
    for (int i = 0; i < 10; ++i) a += h1[i] * sW2[i * 10 + o];
    h2[o] = silu_f(a);
  }

  const float inv_sq3 = 0.5773502691896258f;
  const float inv_sq2 = 0.7071067811865476f;

#pragma unroll
  for (int half = 0; half < 2; ++half) {
    int c = lane + 32 * half;  // channel u in [0,64)
    float w[5];
#pragma unroll
    for (int s5 = 0; s5 < 5; ++s5) {
      float a = sB3[s5 * 64 + c];
#pragma unroll
      for (int i = 0; i < 10; ++i) a += h2[i] * sW3[i * 320 + s5 * 64 + c];
      w[s5] = a;
    }
    float sj = ps[(size_t)j * 64 + c];
    const float* pvj = pv + (size_t)j * 192 + c * 3;
    float vx = pvj[0], vy = pvj[1], vz = pvj[2];

    // scalar messages
    float m0a = w[0] * sj * e0;
    float dot = vx * e1x + vy * e1y + vz * e1z;
    float m0b = w[3] * dot * inv_sq3;
    atomicAddF(&M0[(size_t)n * 128 + c],      m0a);
    atomicAddF(&M0[(size_t)n * 128 + 64 + c], m0b);

    // vector messages
    float crx = vy * e1z - vz * e1y;
    float cry = vz * e1x - vx * e1z;
    float crz = vx * e1y - vy * e1x;
    float* pA = M1o + (size_t)n * 384 + c * 3;
    atomicAddF(pA + 0, w[1] * sj * e1x);
    atomicAddF(pA + 1, w[1] * sj * e1y);
    atomicAddF(pA + 2, w[1] * sj * e1z);
    float* pB = M1o + (size_t)n * 384 + (64 + c) * 3;
    atomicAddF(pB + 0, w[2] * vx * e0);
    atomicAddF(pB + 1, w[2] * vy * e0);
    atomicAddF(pB + 2, w[2] * vz * e0);
    float* pE = M1e + (size_t)n * 192 + c * 3;
    atomicAddF(pE + 0, w[4] * crx * inv_sq2);
    atomicAddF(pE + 1, w[4] * cry * inv_sq2);
    atomicAddF(pE + 2, w[4] * crz * inv_sq2);
  }
}

// ---------------------------------------------------------------------------
// Kernel 3a: g_s = [M0 | onehot-expanded s] @ Wcs  (K=384, Nout=192)
// Species mixing via K-augmentation -> fixed-weight WMMA GEMM.
// nc==0 columns are the scalar output -> SiLU fused, written straight to out.
// ---------------------------------------------------------------------------
__global__ __launch_bounds__(128) void k_post_s(
    const float* __restrict__ M0, const float* __restrict__ emb,
    const int* __restrict__ species, const v2f* __restrict__ Wcs,
    float* __restrict__ gs, float* __restrict__ out, int N, int nTiles) {
  __shared__ v2f Bs2[64 * 64];  // 32KB paired K-chunk of weights
  int t = threadIdx.x, wave = t >> 5, lane = t & 31;
  int tile = blockIdx.x * 4 + wave;
  bool active = tile < nTiles;
  int m = lane & 15, kk = (lane >> 4) << 1, kh = lane >> 4;
  int node = active ? (tile * 16 + m) : 0;
  if (node > N - 1) node = N - 1;
  int sp = species[node];
  const float* m0row = M0 + (size_t)node * 128;
  const float* srow  = emb + (size_t)node * 256;
  int rbase = tile * 16 + (kh << 3);
  bool full = active && (rbase + 8 <= N);

  for (int nc = 0; nc < 3; ++nc) {         // 64-wide output column chunks
    v8f acc[4] = {};
    for (int kc = 0; kc < 3; ++kc) {       // 128-deep K chunks
      const v2f* src = Wcs + (size_t)(nc * 3 + kc) * 4096;
      __syncthreads();
      for (int i = t; i < 4096; i += 128) Bs2[i] = src[i];  // linear b64 copy
      __syncthreads();
      if (active) {
        for (int kt = 0; kt < 32; ++kt) {
          int kglob = kc * 128 + kt * 4 + kk;
          v2f a;
          if (kglob < 128) {               // M0 part
            a.x = m0row[kglob]; a.y = m0row[kglob + 1];
          } else {                          // one-hot expanded s part
            int u = (kglob - 128) >> 2;
            float sv = srow[u];
            a.x = (kk == sp) ? sv : 0.0f;
            a.y = ((kk + 1) == sp) ? sv : 0.0f;
          }
          int pr = kt * 2 + kh;
#pragma unroll
          for (int nt = 0; nt < 4; ++nt) {
            v2f b = Bs2[pr * 64 + nt * 16 + m];
            acc[nt] = wmma4(a, b, acc[nt]);
          }
        }
      }
    }
    if (active) {
      for (int nt = 0; nt < 4; ++nt) {
        int col = nc * 64 + nt * 16 + (lane & 15);
        if (full) {
          if (nc == 0) {
            float* dst = out + (size_t)rbase * 448 + col;
#pragma unroll
            for (int rr = 0; rr < 8; ++rr) dst[(size_t)rr * 448] = silu_f(acc[nt][rr]);
          } else {
            float* dst = gs + (size_t)rbase * 192 + col;
#pragma unroll
            for (int rr = 0; rr < 8; ++rr) dst[(size_t)rr * 192] = acc[nt][rr];
          }
        } else {
          for (int rr = 0; rr < 8; ++rr) {
            int nn = rbase + rr;
            if (nn < N) {
              float val = acc[nt][rr];
              if (nc == 0) out[(size_t)nn * 448 + col] = silu_f(val);
              else         gs[(size_t)nn * 192 + col] = val;
            }
          }
        }
      }
    }
  }
}

// ---------------------------------------------------------------------------
// Kernel 3b: per component c:
//   g_v[:,:,c] = [M1o[:,:,c] | onehot-expanded v[:,:,c]] @ Wcv   (K=384)
//   g_e[:,:,c] = M1e[:,:,c] @ We                                  (K=64)
// SiLU gating fused in the epilogue, final out_v / out_e written directly.
// ---------------------------------------------------------------------------
__global__ __launch_bounds__(128) void k_post_v(
    const float* __restrict__ M1o, const float* __restrict__ M1e,
    const float* __restrict__ emb, const int* __restrict__ species,
    const v2f* __restrict__ Wcv, const v2f* __restrict__ We,
    const float* __restrict__ gs, float* __restrict__ out, int N, int nTiles) {
  __shared__ v2f Bv2[64 * 64];  // 32KB paired K-chunk
  __shared__ v2f Be2[32 * 64];  // 16KB paired
  int t = threadIdx.x, wave = t >> 5, lane = t & 31;
  for (int i = t; i < 32 * 64; i += 128) Be2[i] = We[i];
  int tile = blockIdx.x * 4 + wave;
  bool active = tile < nTiles;
  int m = lane & 15, kk = (lane >> 4) << 1, kh = lane >> 4;
  int node = active ? (tile * 16 + m) : 0;
  if (node > N - 1) node = N - 1;
  int sp = species[node];
  const float* m1row  = M1o + (size_t)node * 384;
  const float* m1erow = M1e + (size_t)node * 192;
  const float* vrow   = emb + (size_t)node * 256 + 64;
  int rbase = tile * 16 + (kh << 3);
  bool full = active && (rbase + 8 <= N);

  for (int c = 0; c < 3; ++c) {
    v8f acc[4] = {};
    for (int kc = 0; kc < 3; ++kc) {
      const v2f* src = Wcv + (size_t)kc * 4096;
      __syncthreads();
      for (int i = t; i < 4096; i += 128) Bv2[i] = src[i];
      __syncthreads();
      if (active) {
        for (int kt = 0; kt < 32; ++kt) {
          int kglob = kc * 128 + kt * 4 + kk;
          v2f a;
          if (kglob < 128) {
            a.x = m1row[kglob * 3 + c]; a.y = m1row[(kglob + 1) * 3 + c];
          } else {
            int u = (kglob - 128) >> 2;
            float vv = vrow[u * 3 + c];
            a.x = (kk == sp) ? vv : 0.0f;
            a.y = ((kk + 1) == sp) ? vv : 0.0f;
          }
          int pr = kt * 2 + kh;
#pragma unroll
          for (int nt = 0; nt < 4; ++nt) {
            v2f b = Bv2[pr * 64 + nt * 16 + m];
            acc[nt] = wmma4(a, b, acc[nt]);
          }
        }
      }
    }
    if (active) {
      // out_v = g_v * silu(gs[:,64:128])
      for (int nt = 0; nt < 4; ++nt) {
        int u = nt * 16 + (lane & 15);
        if (full) {
          const float* gp = gs + (size_t)rbase * 192 + 64 + u;
          float* dst = out + (size_t)rbase * 448 + 64 + u * 3 + c;
#pragma unroll
          for (int rr = 0; rr < 8; ++rr)
            dst[(size_t)rr * 448] = acc[nt][rr] * silu_f(gp[(size_t)rr * 192]);
        } else {
          for (int rr = 0; rr < 8; ++rr) {
            int nn = rbase + rr;
            if (nn < N) {
              float gate = silu_f(gs[(size_t)nn * 192 + 64 + u]);
              out[(size_t)nn * 448 + 64 + u * 3 + c] = acc[nt][rr] * gate;
            }
          }
        }
      }
      // g_e tile + gating: out_e = g_e * silu(gs[:,128:192])
      for (int nt = 0; nt < 4; ++nt) {
        v8f acce = {};
        for (int kt = 0; kt < 16; ++kt) {
          int k = kt * 4 + kk;
          v2f a; a.x = m1erow[k * 3 + c]; a.y = m1erow[(k + 1) * 3 + c];
          v2f b = Be2[(kt * 2 + kh) * 64 + nt * 16 + m];
          acce = wmma4(a, b, acce);
        }
        int u = nt * 16 + (lane & 15);
        if (full) {
          const float* gp = gs + (size_t)rbase * 192 + 128 + u;
          float* dst = out + (size_t)rbase * 448 + 256 + u * 3 + c;
#pragma unroll
          for (int rr = 0; rr < 8; ++rr)
            dst[(size_t)rr * 448] = acce[rr] * silu_f(gp[(size_t)rr * 192]);
        } else {
          for (int rr = 0; rr < 8; ++rr) {
            int nn = rbase + rr;
            if (nn < N) {
              float gate = silu_f(gs[(size_t)nn * 192 + 128 + u]);
              out[(size_t)nn * 448 + 256 + u * 3 + c] = acce[rr] * gate;
            }
          }
        }
      }
    }
  }
}

// ---------------------------------------------------------------------------
// Host launcher
// ---------------------------------------------------------------------------
extern "C" void kernel_launch(void* const* d_in, const int* in_sizes, int n_in,
                              void* d_out, int out_size, void* d_ws, size_t ws_size,
                              hipStream_t stream) {
  const float* emb      = (const float*)d_in[0];   // (N, 256)
  const float* dist     = (const float*)d_in[1];   // (E,)
  const float* eemb     = (const float*)d_in[2];   // (E, 4)
  const int*   species  = (const int*)d_in[3];     // (N,)
  const int*   cidx     = (const int*)d_in[4];     // (E,)
  const int*   nidx     = (const int*)d_in[5];     // (E,)
  const float* pw_s     = (const float*)d_in[6];   // (64,64)
  const float* pw_v     = (const float*)d_in[7];   // (64,64)
  const float* bf       = (const float*)d_in[8];   // (8,)
  const float* W1       = (const float*)d_in[9];   // (8,10)
  const float* b1       = (const float*)d_in[10];  // (10,)
  const float* W2       = (const float*)d_in[11];  // (10,10)
  const float* b2       = (const float*)d_in[12];  // (10,)
  const float* W3       = (const float*)d_in[13];  // (10,320)
  const float* b3       = (const float*)d_in[14];  // (320,)
  const float* post_w_s = (const float*)d_in[15];  // (128,192)
  const float* post_w_v = (const float*)d_in[16];  // (128,64)
  const float* post_w_e = (const float*)d_in[17];  // (64,64)
  const float* si_w_s   = (const float*)d_in[18];  // (64,4,192)
  const float* si_w_v   = (const float*)d_in[19];  // (64,4,64)

  float* out = (float*)d_out;
  float* ws  = (float*)d_ws;

  int N = in_sizes[3];  // species count == N_NODES
  int E = in_sizes[1];  // distances count == N_EDGES

  // Workspace layout (floats); gs aliases ps/pv (dead after edge kernel).
  float* ps  = ws;                        // N*64
  float* pv  = ws + (size_t)N * 64;       // N*192
  float* M0  = ws + (size_t)N * 256;      // N*128
  float* M1o = ws + (size_t)N * 384;      // N*384
  float* M1e = ws + (size_t)N * 768;      // N*192
  v2f*   Wcs = (v2f*)(ws + (size_t)N * 960);        // 36864 v2f
  v2f*   Wcv = Wcs + 36864;                          // 12288 v2f
  v2f*   We  = Wcv + 12288;                          // 2048  v2f
  float* gs  = ws;                        // alias of ps/pv region (N*192)

  // Zero the atomic accumulators (M0|M1o|M1e are contiguous: N*704 floats).
  hipMemsetAsync(M0, 0, (size_t)N * 704 * sizeof(float), stream);

  k_build_w<<<(51200 + 255) / 256, 256, 0, stream>>>(
      post_w_s, post_w_v, post_w_e, si_w_s, si_w_v, Wcs, Wcv, We);

  int nTiles = (N + 15) / 16;
  int nBlk = (nTiles + 3) / 4;

  k_node_proj<<<nBlk, 128, 0, stream>>>(emb, pw_s, pw_v, ps, pv, N, nTiles);

  k_edges<<<(E + 7) / 8, 256, 0, stream>>>(
      dist, eemb, cidx, nidx, bf, W1, b1, W2, b2, W3, b3,
      ps, pv, M0, M1o, M1e, E);

  k_post_s<<<nBlk, 128, 0, stream>>>(M0, emb, species, Wcs, gs, out, N, nTiles);

  k_post_v<<<nBlk, 128, 0, stream>>>(M1o, M1e, emb, species, Wcv, We, gs, out,
                                     N, nTiles);
}